// Model_24292335026755
// MI455X (gfx1250) — compile-verified
//
#include <hip/hip_runtime.h>

#define NB   4
#define Vn   2000
#define E2   (2 * Vn)      // 4000 hyperedges
#define CIN  64
#define H0D  128
#define KNN  32
#define AH   32            // ATTN_HID
#define NT   10
#define LRELU 0.01f
#define LNEPS 1e-5f

typedef __attribute__((ext_vector_type(2))) float v2f;
typedef __attribute__((ext_vector_type(8))) float v8f;

// ---------------------------------------------------------------------------
// squared norms per node
// ---------------------------------------------------------------------------
template <int C>
__global__ void sqnorm_kernel(const float* __restrict__ feat,
                              float* __restrict__ sq) {
  int i = blockIdx.x * blockDim.x + threadIdx.x;
  if (i >= NB * Vn) return;
  const float* p = feat + (size_t)i * C;
  float s = 0.f;
#pragma unroll
  for (int c = 0; c < C; ++c) s += p[c] * p[c];
  sq[i] = s;
}

// ---------------------------------------------------------------------------
// kNN: Gram tile via f32 WMMA, 16 full distance rows live in LDS (129KB),
// then 32 rounds of wave argmin per row.  key = |x_v|^2 - 2 x_e.x_v
// C is a template param: C=64 -> unrolled, guard-free, b64 loads;
// C=3 -> single K-step with compile-time masking.
// ---------------------------------------------------------------------------
template <int C>
__global__ __launch_bounds__(256)
void knn_topk_kernel(const float* __restrict__ feat, const float* __restrict__ sq,
                     int* __restrict__ idx_out, int* __restrict__ deg,
                     int edge_base) {
  extern __shared__ float sdist[];              // [16][2016]
  constexpr int KSTEPS = (C + 3) / 4;
  const int n    = blockIdx.y;
  const int row0 = blockIdx.x * 16;
  const int tid  = threadIdx.x;
  const int wave = tid >> 5, lane = tid & 31;
  const int mlo  = lane & 15, kh = lane >> 4;
  const float* fb = feat + (size_t)n * Vn * C;

  // preload A fragments (16 rows x C), layout per ISA 16x4 f32 A-matrix:
  // lane halves hold K pairs {4kk+2kh, 4kk+2kh+1} for row mlo.
  v2f af[KSTEPS];
  if constexpr (C % 4 == 0) {
    const v2f* fr = (const v2f*)(fb + (size_t)(row0 + mlo) * C);
#pragma unroll
    for (int kk = 0; kk < KSTEPS; ++kk) af[kk] = fr[kk * 2 + kh];
  } else {                                      // C == 3
    const float* fr = fb + (size_t)(row0 + mlo) * C;
    af[0].x = fr[2 * kh];
    af[0].y = (kh == 0) ? fr[1] : 0.f;
  }

  for (int ct = wave; ct < 125; ct += 8) {      // 125 column tiles of 16
    int col0 = ct * 16;
    v8f acc = {};
    if constexpr (C % 4 == 0) {
      const v2f* fc = (const v2f*)(fb + (size_t)(col0 + mlo) * C);
#pragma unroll
      for (int kk = 0; kk < KSTEPS; ++kk) {
        v2f b = fc[kk * 2 + kh];
        acc = __builtin_amdgcn_wmma_f32_16x16x4_f32(false, af[kk], false, b,
                                                    (short)0, acc, false, false);
      }
    } else {
      const float* fc = fb + (size_t)(col0 + mlo) * C;
      v2f b;
      b.x = fc[2 * kh];
      b.y = (kh == 0) ? fc[1] : 0.f;
      acc = __builtin_amdgcn_wmma_f32_16x16x4_f32(false, af[0], false, b,
                                                  (short)0, acc, false, false);
    }
    float sqc = sq[n * Vn + col0 + mlo];
#pragma unroll
    for (int r = 0; r < 8; ++r) {
      int m = r + 8 * kh;                       // C/D layout: vgpr r -> M=r / r+8
      sdist[m * 2016 + col0 + mlo] = sqc - 2.f * acc[r];
    }
  }
  __syncthreads();

  // per-row top-K (wave handles rows wave and wave+8)
  for (int half = 0; half < 2; ++half) {
    int m = wave + 8 * half;
    int e = edge_base + row0 + m;
    float* drow = &sdist[m * 2016];
    for (int k = 0; k < KNN; ++k) {
      float best = 3.4e38f; int bi = 0x7fffffff;
      for (int c = lane; c < Vn; c += 32) {
        float vv = drow[c];
        if (vv < best) { best = vv; bi = c; }
      }
      for (int off = 16; off > 0; off >>= 1) {
        float ov = __shfl_down(best, off, 32);
        int   oi = __shfl_down(bi,   off, 32);
        if (ov < best || (ov == best && oi < bi)) { best = ov; bi = oi; }
      }
      bi = __shfl(bi, 0, 32);
      if (lane == 0) {
        idx_out[((size_t)n * E2 + e) * KNN + k] = bi;
        atomicAdd(&deg[n * Vn + bi], 1);
        drow[bi] = 3.4e38f;
      }
      __builtin_amdgcn_wave_barrier();
    }
  }
}

// ---------------------------------------------------------------------------
// CSR build: exclusive prefix over degrees, then fill node->edge lists
// ---------------------------------------------------------------------------
__global__ void prefix_kernel(const int* __restrict__ deg, int* __restrict__ off) {
  int n = blockIdx.x;
  int run = 0;
  for (int v = 0; v < Vn; ++v) { off[n * Vn + v] = run; run += deg[n * Vn + v]; }
}

__global__ void csr_fill_kernel(const int* __restrict__ idx, const int* __restrict__ off,
                                int* __restrict__ cursor, int* __restrict__ csr_e) {
  int t = blockIdx.x * blockDim.x + threadIdx.x;
  if (t >= NB * E2 * KNN) return;
  int n   = t / (E2 * KNN);
  int rem = t - n * (E2 * KNN);
  int e   = rem / KNN;
  int v   = idx[t];
  int pos = atomicAdd(&cursor[n * Vn + v], 1);
  csr_e[(size_t)n * (E2 * KNN) + off[n * Vn + v] + pos] = e;
}

// ---------------------------------------------------------------------------
// input projections: y0 = x @ l1_theta ; res = x @ res_theta + res_bias (WMMA)
// ---------------------------------------------------------------------------
__global__ __launch_bounds__(256)
void input_proj_kernel(const float* __restrict__ x, const float* __restrict__ Wl1,
                       const float* __restrict__ Wres, const float* __restrict__ res_b,
                       float* __restrict__ y0, float* __restrict__ res) {
  __shared__ __align__(16) float sA[16 * CIN];
  const int row0 = blockIdx.x * 16;             // 500 tiles over 8000 rows
  const int tid  = threadIdx.x, wave = tid >> 5, lane = tid & 31;
  const int mlo  = lane & 15, kh = lane >> 4;
  for (int i = tid; i < 16 * CIN; i += 256) sA[i] = x[(size_t)row0 * CIN + i];
  __syncthreads();
  const v2f* sA2 = (const v2f*)sA;
  for (int task = wave; task < 16; task += 8) { // 2 outputs x 8 col-tiles
    const float* W = (task < 8) ? Wl1 : Wres;
    int col0 = (task & 7) * 16;
    v8f acc = {};
#pragma unroll
    for (int kk = 0; kk < CIN / 4; ++kk) {
      int k0 = kk * 4 + 2 * kh;
      v2f a = sA2[mlo * (CIN / 2) + kk * 2 + kh];   // ds_load_b64
      v2f b;
      b.x = W[k0 * H0D + col0 + mlo];
      b.y = W[(k0 + 1) * H0D + col0 + mlo];
      acc = __builtin_amdgcn_wmma_f32_16x16x4_f32(false, a, false, b,
                                                  (short)0, acc, false, false);
    }
    float* out = (task < 8) ? y0 : res;
    float  bb  = (task < 8) ? 0.f : res_b[col0 + mlo];
#pragma unroll
    for (int r = 0; r < 8; ++r) {
      int row = row0 + r + 8 * kh;
      out[(size_t)row * H0D + col0 + mlo] = acc[r] + bb;
    }
  }
}

// ---------------------------------------------------------------------------
// sparse hyconv: edge aggregation (mean of 32 members) then node aggregation
// ---------------------------------------------------------------------------
__global__ __launch_bounds__(128)
void edge_gather_kernel(const float* __restrict__ h, const int* __restrict__ idx,
                        float* __restrict__ ebuf) {
  int e = blockIdx.x, n = blockIdx.y, c = threadIdx.x;
  __shared__ int sidx[KNN];
  if (c < KNN) sidx[c] = idx[((size_t)n * E2 + e) * KNN + c];
  __syncthreads();
  const float* hb = h + (size_t)n * Vn * H0D;
  float acc = 0.f;
#pragma unroll 8
  for (int k = 0; k < KNN; ++k) {
    if (k + 8 < KNN)                            // global_prefetch_b8 ahead
      __builtin_prefetch(&hb[(size_t)sidx[k + 8] * H0D + c], 0, 1);
    acc += hb[(size_t)sidx[k] * H0D + c];
  }
  ebuf[((size_t)n * E2 + e) * H0D + c] = acc * (1.f / KNN);
}

// out = a_keep*hprev + a_new*(sum_e ebuf)/deg + bias
__global__ __launch_bounds__(128)
void node_gather_kernel(const float* __restrict__ ebuf, const int* __restrict__ csr_e,
                        const int* __restrict__ off, const int* __restrict__ deg,
                        const float* __restrict__ hprev, const float* __restrict__ bias,
                        float a_keep, float a_new, float* __restrict__ out) {
  int v = blockIdx.x, n = blockIdx.y, c = threadIdx.x;
  int d = deg[n * Vn + v], o = off[n * Vn + v];
  const int*   ce = csr_e + (size_t)n * (E2 * KNN) + o;
  const float* eb = ebuf + (size_t)n * E2 * H0D;
  float acc = 0.f;
  for (int j = 0; j < d; ++j) acc += eb[(size_t)ce[j] * H0D + c];
  float val = a_new * (acc / (float)d);
  if (bias)  val += bias[c];
  size_t i = ((size_t)n * Vn + v) * H0D + c;
  if (hprev) val += a_keep * hprev[i];
  out[i] = val;
}

// ---------------------------------------------------------------------------
// h = leaky(LN(y + res))
// ---------------------------------------------------------------------------
__global__ __launch_bounds__(128)
void ln_leaky_kernel(const float* __restrict__ y, const float* __restrict__ res,
                     const float* __restrict__ g, const float* __restrict__ b,
                     float* __restrict__ out) {
  int v = blockIdx.x, n = blockIdx.y, c = threadIdx.x;
  size_t i = ((size_t)n * Vn + v) * H0D + c;
  float t = y[i] + res[i];
  __shared__ float red[H0D];
  red[c] = t; __syncthreads();
  for (int s = 64; s > 0; s >>= 1) { if (c < s) red[c] += red[c + s]; __syncthreads(); }
  float mu = red[0] * (1.f / H0D); __syncthreads();
  float d0 = t - mu;
  red[c] = d0 * d0; __syncthreads();
  for (int s = 64; s > 0; s >>= 1) { if (c < s) red[c] += red[c + s]; __syncthreads(); }
  float var = red[0] * (1.f / H0D);
  float r = (t - mu) * rsqrtf(var + LNEPS) * g[c] + b[c];
  out[i] = (r > 0.f) ? r : LRELU * r;
}

// ---------------------------------------------------------------------------
// HDS dense layer: out = h + leaky(h @ W + b)   (WMMA, K=128)
// ---------------------------------------------------------------------------
__global__ __launch_bounds__(256)
void hds_dense_kernel(const float* __restrict__ h, const float* __restrict__ W,
                      const float* __restrict__ bias, float* __restrict__ out) {
  __shared__ __align__(16) float sA[16 * H0D];
  const int row0 = blockIdx.x * 16;
  const int tid  = threadIdx.x, wave = tid >> 5, lane = tid & 31;
  const int mlo  = lane & 15, kh = lane >> 4;
  for (int i = tid; i < 16 * H0D; i += 256) sA[i] = h[(size_t)row0 * H0D + i];
  __syncthreads();
  const v2f* sA2 = (const v2f*)sA;
  int col0 = wave * 16;
  v8f acc = {};
#pragma unroll
  for (int kk = 0; kk < H0D / 4; ++kk) {
    int k0 = kk * 4 + 2 * kh;
    v2f a = sA2[mlo * (H0D / 2) + kk * 2 + kh];     // ds_load_b64
    v2f b;
    b.x = W[k0 * H0D + col0 + mlo];
    b.y = W[(k0 + 1) * H0D + col0 + mlo];
    acc = __builtin_amdgcn_wmma_f32_16x16x4_f32(false, a, false, b,
                                                (short)0, acc, false, false);
  }
  float bb = bias[col0 + mlo];
#pragma unroll
  for (int r = 0; r < 8; ++r) {
    int row = row0 + r + 8 * kh;
    float t = acc[r] + bb;
    t = (t > 0.f) ? t : LRELU * t;
    size_t i = (size_t)row * H0D + col0 + mlo;
    out[i] = h[i] + t;
  }
}

// ---------------------------------------------------------------------------
// attention, rearranged:  ha = sum_v a_v h_v ; m = Wq (Wk^T ha);
// attn_v = scale * h_v . m ; L1-normalize ; pooled ; logits
// ---------------------------------------------------------------------------
__global__ __launch_bounds__(128)
void ha_reduce_kernel(const float* __restrict__ h, const float* __restrict__ a,
                      float* __restrict__ ha) {
  int n = blockIdx.x, c = threadIdx.x;
  const float* hb = h + (size_t)n * Vn * H0D;
  float acc = 0.f;
  for (int v = 0; v < Vn; ++v) acc += a[v] * hb[(size_t)v * H0D + c];
  ha[n * H0D + c] = acc;
}

__global__ __launch_bounds__(128)
void attn_vec_kernel(const float* __restrict__ ha, const float* __restrict__ Wq,
                     const float* __restrict__ Wk, float* __restrict__ mvec) {
  int n = blockIdx.x, c = threadIdx.x;
  __shared__ float kv[AH];
  if (c < AH) {
    float acc = 0.f;
    for (int cc = 0; cc < H0D; ++cc) acc += ha[n * H0D + cc] * Wk[cc * AH + c];
    kv[c] = acc;
  }
  __syncthreads();
  float acc = 0.f;
#pragma unroll
  for (int j = 0; j < AH; ++j) acc += Wq[c * AH + j] * kv[j];
  mvec[n * H0D + c] = acc;
}

__global__ __launch_bounds__(128)
void attn_score_kernel(const float* __restrict__ h, const float* __restrict__ mvec,
                       float* __restrict__ attn, float* __restrict__ asum) {
  int v = blockIdx.x, n = blockIdx.y, c = threadIdx.x;
  __shared__ float red[H0D];
  red[c] = h[((size_t)n * Vn + v) * H0D + c] * mvec[n * H0D + c];
  __syncthreads();
  for (int s = 64; s > 0; s >>= 1) { if (c < s) red[c] += red[c + s]; __syncthreads(); }
  if (c == 0) {
    float sc = red[0] * 0.17677669529663687f;   // 32^-0.5
    attn[n * Vn + v] = sc;
    atomicAdd(&asum[n], fabsf(sc));
  }
}

__global__ __launch_bounds__(128)
void pool_logits_kernel(const float* __restrict__ h, const float* __restrict__ attn,
                        const float* __restrict__ asum, const float* __restrict__ fcw,
                        const float* __restrict__ fcb, float* __restrict__ d_out) {
  int n = blockIdx.x, c = threadIdx.x;
  float inv = 1.f / fmaxf(asum[n], 1e-12f);
  const float* hb = h + (size_t)n * Vn * H0D;
  float acc = 0.f;
  for (int v = 0; v < Vn; ++v) acc += attn[n * Vn + v] * hb[(size_t)v * H0D + c];
  acc *= inv;
  __shared__ float pooled[H0D];
  pooled[c] = acc;
  d_out[NB * NT + n * H0D + c] = acc;           // pooled after logits
  __syncthreads();
  if (c < NT) {
    float l = fcb[c];
    for (int cc = 0; cc < H0D; ++cc) l += pooled[cc] * fcw[cc * NT + c];
    d_out[n * NT + c] = l;
  }
}

// ---------------------------------------------------------------------------
extern "C" void kernel_launch(void* const* d_in, const int* in_sizes, int n_in,
                              void* d_out, int out_size, void* d_ws, size_t ws_size,
                              hipStream_t stream) {
  const float* x        = (const float*)d_in[0];
  const float* coors    = (const float*)d_in[1];
  const float* l1_theta = (const float*)d_in[2];
  const float* l1_bias  = (const float*)d_in[3];
  const float* l1_ln_g  = (const float*)d_in[4];
  const float* l1_ln_b  = (const float*)d_in[5];
  const float* res_theta= (const float*)d_in[6];
  const float* res_bias = (const float*)d_in[7];
  const float* hds_w    = (const float*)d_in[8];
  const float* hds_b    = (const float*)d_in[9];
  const float* attn_q   = (const float*)d_in[10];
  const float* attn_k   = (const float*)d_in[11];
  const float* attn_a   = (const float*)d_in[12];
  const float* fc_w     = (const float*)d_in[13];
  const float* fc_b     = (const float*)d_in[14];

  char* ws = (char*)d_ws;
  size_t o = 0;
  auto carve = [&](size_t bytes) -> void* {
    void* p = ws + o; o += (bytes + 255) & ~(size_t)255; return p;
  };
  float* sqx   = (float*)carve((size_t)NB * Vn * 4);
  float* sqc   = (float*)carve((size_t)NB * Vn * 4);
  int*   idx   = (int*)  carve((size_t)NB * E2 * KNN * 4);
  int*   deg   = (int*)  carve((size_t)NB * Vn * 4);
  int*   off   = (int*)  carve((size_t)NB * Vn * 4);
  int*   cur_c = (int*)  carve((size_t)NB * Vn * 4);
  int*   csr   = (int*)  carve((size_t)NB * E2 * KNN * 4);
  float* y0    = (float*)carve((size_t)NB * Vn * H0D * 4);
  float* resb  = (float*)carve((size_t)NB * Vn * H0D * 4);
  float* ebuf  = (float*)carve((size_t)NB * E2 * H0D * 4);
  float* hA    = (float*)carve((size_t)NB * Vn * H0D * 4);
  float* hB    = (float*)carve((size_t)NB * Vn * H0D * 4);
  float* ha    = (float*)carve((size_t)NB * H0D * 4);
  float* mvec  = (float*)carve((size_t)NB * H0D * 4);
  float* attn  = (float*)carve((size_t)NB * Vn * 4);
  float* asum  = (float*)carve((size_t)NB * 4);

  hipMemsetAsync(deg,   0, (size_t)NB * Vn * 4, stream);
  hipMemsetAsync(cur_c, 0, (size_t)NB * Vn * 4, stream);
  hipMemsetAsync(asum,  0, (size_t)NB * 4, stream);

  sqnorm_kernel<CIN><<<(NB * Vn + 255) / 256, 256, 0, stream>>>(x, sqx);
  sqnorm_kernel<3>  <<<(NB * Vn + 255) / 256, 256, 0, stream>>>(coors, sqc);

  dim3 kgrid(125, NB);
  size_t shl = (size_t)16 * 2016 * 4;           // 129 KB LDS distance rows
  knn_topk_kernel<CIN><<<kgrid, 256, shl, stream>>>(x,     sqx, idx, deg, 0);
  knn_topk_kernel<3>  <<<kgrid, 256, shl, stream>>>(coors, sqc, idx, deg, Vn);

  prefix_kernel<<<NB, 1, 0, stream>>>(deg, off);
  csr_fill_kernel<<<(NB * E2 * KNN + 255) / 256, 256, 0, stream>>>(idx, off, cur_c, csr);

  input_proj_kernel<<<(NB * Vn) / 16, 256, 0, stream>>>(x, l1_theta, res_theta,
                                                        res_bias, y0, resb);

  dim3 eg(E2, NB), ng(Vn, NB);
  edge_gather_kernel<<<eg, 128, 0, stream>>>(y0, idx, ebuf);
  node_gather_kernel<<<ng, 128, 0, stream>>>(ebuf, csr, off, deg,
                                             nullptr, l1_bias, 0.f, 1.f, hA);
  ln_leaky_kernel<<<ng, 128, 0, stream>>>(hA, resb, l1_ln_g, l1_ln_b, hA);

  float* cur = hA; float* oth = hB;
  for (int layer = 0; layer < 2; ++layer) {
    hds_dense_kernel<<<(NB * Vn) / 16, 256, 0, stream>>>(cur, hds_w, hds_b, oth);
    { float* t = cur; cur = oth; oth = t; }
    for (int s = 0; s < 10; ++s) {              // h = 0.95 h + 0.05 hyconv(h)
      edge_gather_kernel<<<eg, 128, 0, stream>>>(cur, idx, ebuf);
      node_gather_kernel<<<ng, 128, 0, stream>>>(ebuf, csr, off, deg,
                                                 cur, nullptr, 0.95f, 0.05f, oth);
      float* t = cur; cur = oth; oth = t;
    }
  }

  ha_reduce_kernel<<<NB, 128, 0, stream>>>(cur, attn_a, ha);
  attn_vec_kernel<<<NB, 128, 0, stream>>>(ha, attn_q, attn_k, mvec);
  attn_score_kernel<<<ng, 128, 0, stream>>>(cur, mvec, attn, asum);
  pool_logits_kernel<<<NB, 128, 0, stream>>>(cur, attn, asum, fc_w, fc_b,
                                             (float*)d_out);
}